// ood_mahalanobis_loss_76209899700427
// MI455X (gfx1250) — compile-verified
//
#include <hip/hip_runtime.h>

// ---------------------------------------------------------------------------
// MI455X (gfx1250, wave32) Mahalanobis OOD loss. B=2048, G=64, D=512.
//   K1: blocked Gauss-Jordan inverse of 64 SPD 512x512 matrices.
//       Panels (~200KB) live in LDS; trailing rank-32 update is an fp32 WMMA
//       GEMM -> matrix L2 traffic drops from 512 passes to 16 (~2.2GB).
//   K2: dist[b,g] = diff^T inv_cov diff, 68.7 GFLOP fp32 via
//       V_WMMA_F32_16X16X4_F32; 128-row tiles, 8 waves (2 waves/SIMD),
//       next B-panel prefetched into VGPRs while current slab computes.
//   K3: relu -> min_g -> mean_b -> -x/1e4.
// HBM traffic ~130MB (5.6us @ 23.3TB/s) vs ~100 GFLOP fp32 -> compute-bound,
// so everything hot runs on the native CDNA5 fp32 WMMA path.
// ---------------------------------------------------------------------------

typedef float v2f __attribute__((ext_vector_type(2)));
typedef float v8f __attribute__((ext_vector_type(8)));

#define BB 2048
#define GG 64
#define DD 512
#define PW 32   // GJ panel width

// ---------------------------------------------------------------------------
// Kernel 1: blocked in-place Gauss-Jordan inversion. One 256-thread block
// (8 waves) per matrix. Per outer step (16 total):
//   A: stage column panel M[:,S] and pivot rows M[S,:] (live + old) in LDS
//   B: 32 sequential micro-GJ pivots entirely in LDS
//   C: trailing update M_new = M_old + colP_final x rowP_old via fp32 WMMA
//      (K=32, accumulator starts at 0 so the M_old tile loads overlap the
//      WMMA chain and are folded in with a vector add at the end);
//      batch-row/batch-col tiles are written straight from the panels
//      (tile-uniform scalar branches).
// WMMA operand layouts per CDNA5 ISA 7.12.2 (f32 16x16x4):
//   A: lane=M row, a[v]=A[M][K=2*half+v]; B: b[v]=B[K=2*half+v][N=lane%16];
//   C/D: c[v]=C[M=v+8*half][N=lane%16].
// ---------------------------------------------------------------------------
__global__ __launch_bounds__(256) void gj_invert_blocked_kernel(
    const float* __restrict__ covs, float* __restrict__ inv) {
  __shared__ float colP[DD][34];       // column panel; stride 34 -> conflict-free A frags
  __shared__ float rowP[PW][520];      // pivot rows (live); stride 520 -> halves disjoint
  __shared__ float rowPold[PW][520];   // pivot rows (pre-batch) = B operand
  __shared__ float fst[DD];            // per-row multipliers for one micro-pivot

  const int g    = blockIdx.x;
  const int tid  = threadIdx.x;
  const int lane = tid & 31;
  const int h    = (lane >> 4) & 1;
  const int l16  = lane & 15;
  // Force the wave id scalar so the phase-C tile loop + classification are
  // uniform (s_cbranch, EXEC stays all-ones for every executed WMMA).
  const int swave = __builtin_amdgcn_readfirstlane(tid >> 5);   // 0..7

  const float* C = covs + (size_t)g * DD * DD;
  float*       M = inv  + (size_t)g * DD * DD;

  {  // vectorized 1MB copy covs -> workspace (b128)
    const float4* C4 = (const float4*)C;
    float4*       M4 = (float4*)M;
    for (int idx = tid; idx < DD * DD / 4; idx += 256) M4[idx] = C4[idx];
  }
  __syncthreads();

  for (int k0 = 0; k0 < DD; k0 += PW) {
    // ---- Phase A: stage panels ----
    for (int idx = tid; idx < DD * PW; idx += 256) {
      const int i = idx >> 5, c = idx & (PW - 1);
      colP[i][c] = M[(size_t)i * DD + (k0 + c)];
    }
    for (int idx = tid; idx < PW * (DD / 4); idx += 256) {   // b128 rows
      const int r = idx >> 7, c4 = (idx & 127) << 2;
      const float4 v = *(const float4*)(M + (size_t)(k0 + r) * DD + c4);
      *(float4*)&rowP[r][c4]    = v;
      *(float4*)&rowPold[r][c4] = v;
    }
    __syncthreads();

    // ---- Phase B: micro Gauss-Jordan within the panels ----
    for (int kk = 0; kk < PW; ++kk) {
      const int k = k0 + kk;
      const float pv = rowP[kk][k];                    // pivot (pre-scale)
      for (int i = tid; i < DD; i += 256) fst[i] = colP[i][kk];
      __syncthreads();
      const float pinv = 1.0f / pv;
      // scale pivot row (in-place rule: A[k][k] := 1 first); mirror into colP
      for (int j = tid; j < DD; j += 256) {
        const float val = ((j == k) ? 1.0f : rowP[kk][j]) * pinv;
        rowP[kk][j] = val;
        if ((unsigned)(j - k0) < (unsigned)PW) colP[k][j - k0] = val;
      }
      __syncthreads();
      // eliminate inside column panel (all rows i != k)
      for (int i = tid; i < DD; i += 256) {
        if (i == k) continue;
        const float f = fst[i];
        #pragma unroll
        for (int c = 0; c < PW; ++c) {
          const float old = (c == kk) ? 0.0f : colP[i][c];
          colP[i][c] = fmaf(-f, rowP[kk][k0 + c], old);
        }
      }
      // eliminate inside pivot-row panel (rows ii != kk, full width)
      for (int idx = tid; idx < PW * DD; idx += 256) {
        const int ii = idx >> 9, j = idx & (DD - 1);
        if (ii == kk) continue;
        const float f   = fst[k0 + ii];
        const float old = (j == k) ? 0.0f : rowP[ii][j];
        rowP[ii][j] = fmaf(-f, rowP[kk][j], old);
      }
      __syncthreads();
    }

    // ---- Phase C: trailing rank-32 update via fp32 WMMA + write-back ----
    for (int t = swave; t < (DD / 16) * (DD / 16); t += 8) {
      const int r0 = (t >> 5) << 4;
      const int c0 = (t & 31) << 4;
      const bool rowInS = (unsigned)(r0 - k0) < (unsigned)PW;  // whole tile in S
      const bool colInS = (unsigned)(c0 - k0) < (unsigned)PW;
      if (rowInS) {
        #pragma unroll
        for (int v = 0; v < 8; ++v) {
          const int grow = r0 + v + 8 * h;
          M[(size_t)grow * DD + (c0 + l16)] = rowP[grow - k0][c0 + l16];
        }
      } else if (colInS) {
        #pragma unroll
        for (int v = 0; v < 8; ++v) {
          const int grow = r0 + v + 8 * h;
          M[(size_t)grow * DD + (c0 + l16)] = colP[grow][(c0 + l16) - k0];
        }
      } else {
        // issue M_old tile loads; they complete under the WMMA chain
        float cold[8];
        #pragma unroll
        for (int v = 0; v < 8; ++v)
          cold[v] = M[(size_t)(r0 + v + 8 * h) * DD + (c0 + l16)];
        v8f acc = {};
        #pragma unroll
        for (int ks = 0; ks < PW; ks += 4) {
          const int ka = ks + 2 * h;
          v2f a, b;
          a.x = colP[r0 + l16][ka];
          a.y = colP[r0 + l16][ka + 1];
          b.x = rowPold[ka][c0 + l16];
          b.y = rowPold[ka + 1][c0 + l16];
          acc = __builtin_amdgcn_wmma_f32_16x16x4_f32(false, a, false, b,
                                                      (short)0, acc, false, false);
        }
        #pragma unroll
        for (int v = 0; v < 8; ++v)
          M[(size_t)(r0 + v + 8 * h) * DD + (c0 + l16)] = acc[v] + cold[v];
      }
    }
    __syncthreads();
  }
}

// ---------------------------------------------------------------------------
// Kernel 2: fused diff / GEMM / quadratic-form via fp32 WMMA.
// Grid: (B/128, G). Block: 256 threads = 8 waves; wave w owns rows w*16..+15.
// diff tile 128x512 (stride 516: pad-4 -> conflict-free b64 A loads, halves
// on disjoint bank pairs). B panels are register-prefetched one slab ahead:
// the 32 global loads for slab jt+1 are in flight during slab jt's 128 WMMAs per wave.
// ---------------------------------------------------------------------------
__global__ __launch_bounds__(256) void mahal_wmma_kernel(
    const float* __restrict__ X, const float* __restrict__ mus,
    const float* __restrict__ invc, float* __restrict__ dist) {
  constexpr int TB   = 128;
  constexpr int PAD  = 4;
  constexpr int NPF  = (DD * 16) / 256;   // 32 panel elements per thread
  __shared__ float diffL[TB][DD + PAD];   // 258 KB
  __shared__ float panelL[DD * 16];       // 32 KB

  const int g    = blockIdx.y;
  const int b0   = blockIdx.x * TB;
  const int tid  = threadIdx.x;
  const int lane = tid & 31;
  const int wave = tid >> 5;              // 0..7
  const int h    = (lane >> 4) & 1;
  const int l16  = lane & 15;

  const float* mu = mus  + (size_t)g * DD;
  const float* Mg = invc + (size_t)g * DD * DD;

  // Stage diff = x - mu, vectorized b128 (rows 16B aligned: stride 516, c%4==0).
  for (int idx = tid; idx < TB * (DD / 4); idx += 256) {
    const int r  = idx >> 7;
    const int c4 = (idx & 127) << 2;
    const float4 xv = *(const float4*)(X + (size_t)(b0 + r) * DD + c4);
    const float4 mv = *(const float4*)(mu + c4);
    float4 d;
    d.x = xv.x - mv.x; d.y = xv.y - mv.y; d.z = xv.z - mv.z; d.w = xv.w - mv.w;
    *(float4*)&diffL[r][c4] = d;
  }

  float acc[8];
  #pragma unroll
  for (int v = 0; v < 8; ++v) acc[v] = 0.0f;

  const int arow = wave * 16 + l16;

  // Prefetch registers for the B panel (inv_cov[:, j0:j0+16], [k][n] layout).
  float pf[NPF];
  #pragma unroll
  for (int i = 0; i < NPF; ++i) {
    const int idx = tid + i * 256;
    pf[i] = Mg[(size_t)(idx >> 4) * DD + (idx & 15)];     // slab jt = 0
  }

  for (int jt = 0; jt < DD / 16; ++jt) {
    __syncthreads();                      // previous panel fully consumed
    #pragma unroll
    for (int i = 0; i < NPF; ++i)
      panelL[tid + i * 256] = pf[i];
    __syncthreads();

    if (jt + 1 < DD / 16) {               // issue next slab's loads now;
      const int j0n = (jt + 1) * 16;      // they complete under the WMMAs
      #pragma unroll
      for (int i = 0; i < NPF; ++i) {
        const int idx = tid + i * 256;
        pf[i] = Mg[(size_t)(idx >> 4) * DD + (j0n + (idx & 15))];
      }
    }

    const int j0 = jt * 16;
    v8f c = {};
    #pragma unroll 8
    for (int k0 = 0; k0 < DD; k0 += 4) {
      const int ka = k0 + 2 * h;
      v2f a, b;
      a.x = diffL[arow][ka];
      a.y = diffL[arow][ka + 1];
      b.x = panelL[ka * 16 + l16];
      b.y = panelL[(ka + 1) * 16 + l16];
      c = __builtin_amdgcn_wmma_f32_16x16x4_f32(false, a, false, b,
                                                (short)0, c, false, false);
    }
    #pragma unroll
    for (int v = 0; v < 8; ++v) {
      const int row = wave * 16 + v + 8 * h;
      acc[v] = fmaf(c[v], diffL[row][j0 + l16], acc[v]);
    }
  }

  #pragma unroll
  for (int v = 0; v < 8; ++v) {
    float s = acc[v];
    s += __shfl_xor(s, 1);
    s += __shfl_xor(s, 2);
    s += __shfl_xor(s, 4);
    s += __shfl_xor(s, 8);
    if (l16 == 0) {
      const int row = wave * 16 + v + 8 * h;
      dist[(size_t)(b0 + row) * GG + g] = fmaxf(s, 0.0f);   // relu
    }
  }
}

// ---------------------------------------------------------------------------
// Kernel 3: min over g, mean over b, final scale.
// ---------------------------------------------------------------------------
__global__ __launch_bounds__(256) void minmean_kernel(
    const float* __restrict__ dist, float* __restrict__ out) {
  __shared__ float sdata[256];
  float local = 0.0f;
  for (int b = threadIdx.x; b < BB; b += 256) {
    const float4* row4 = (const float4*)(dist + (size_t)b * GG);
    float4 m4 = row4[0];
    #pragma unroll
    for (int q = 1; q < GG / 4; ++q) {
      const float4 v = row4[q];
      m4.x = fminf(m4.x, v.x); m4.y = fminf(m4.y, v.y);
      m4.z = fminf(m4.z, v.z); m4.w = fminf(m4.w, v.w);
    }
    local += fminf(fminf(m4.x, m4.y), fminf(m4.z, m4.w));
  }
  sdata[threadIdx.x] = local;
  __syncthreads();
  for (int s = 128; s > 0; s >>= 1) {
    if (threadIdx.x < s) sdata[threadIdx.x] += sdata[threadIdx.x + s];
    __syncthreads();
  }
  if (threadIdx.x == 0)
    out[0] = -(sdata[0] / (float)BB) / 10000.0f;
}

// ---------------------------------------------------------------------------
extern "C" void kernel_launch(void* const* d_in, const int* in_sizes, int n_in,
                              void* d_out, int out_size, void* d_ws, size_t ws_size,
                              hipStream_t stream) {
  const float* X    = (const float*)d_in[0];   // [B, D]
  const float* mus  = (const float*)d_in[1];   // [G, D]
  const float* covs = (const float*)d_in[2];   // [G, D, D]

  // Workspace: [0, 64MB) inv_covs ; [64MB, +512KB) dist matrix.
  float* inv  = (float*)d_ws;
  float* dist = (float*)((char*)d_ws + (size_t)GG * DD * DD * sizeof(float));

  gj_invert_blocked_kernel<<<GG, 256, 0, stream>>>(covs, inv);

  dim3 grid(BB / 128, GG);
  mahal_wmma_kernel<<<grid, 256, 0, stream>>>(X, mus, inv, dist);

  minmean_kernel<<<1, 256, 0, stream>>>(dist, (float*)d_out);
}